// GCN_32942399160959
// MI455X (gfx1250) — compile-verified
//
#include <hip/hip_runtime.h>

typedef __attribute__((ext_vector_type(2))) float v2f;
typedef __attribute__((ext_vector_type(8))) float v8f;

#define F_IN 128
#define HID  64
#define NCLS 16

// ---------------- elementwise helpers ----------------

__global__ void zero_f32(float* p, int n) {
    int i = blockIdx.x * blockDim.x + threadIdx.x;
    if (i < n) p[i] = 0.0f;
}

// out[i] = b[i & mask]  (broadcast bias row; mask = C-1, C power of two)
__global__ void bias_broadcast(float* p, const float* b, int n, int mask) {
    int i = blockIdx.x * blockDim.x + threadIdx.x;
    if (i < n) p[i] = b[i & mask];
}

__global__ void relu_inplace(float* p, int n) {
    int i = blockIdx.x * blockDim.x + threadIdx.x;
    if (i < n) p[i] = fmaxf(p[i], 0.0f);
}

// ---------------- degree / norm ----------------

__global__ void deg_count(const int* __restrict__ dst, float* __restrict__ deg, int E) {
    int e = blockIdx.x * blockDim.x + threadIdx.x;
    if (e < E) atomicAdd(&deg[dst[e]], 1.0f);
}

// dis[i] = deg>0 ? rsqrt(max(deg,1)) : 0   (in place)
__global__ void deg_to_dis(float* __restrict__ d, int n) {
    int i = blockIdx.x * blockDim.x + threadIdx.x;
    if (i < n) {
        float v = d[i];
        d[i] = (v > 0.0f) ? rsqrtf(fmaxf(v, 1.0f)) : 0.0f;
    }
}

// ---------------- WMMA GEMM 1: H[N,64] = X[N,128] @ W1[128,64] ----------------
// one wave per 16-row M tile; 4 accumulators cover N=0..63; fp32 WMMA 16x16x4.

__global__ __launch_bounds__(32) void gemm1_wmma(const float* __restrict__ X,
                                                 const float* __restrict__ W1,
                                                 float* __restrict__ H) {
    const int tile = blockIdx.x;          // M tile index
    const int lane = threadIdx.x;         // 0..31
    const int half = lane >> 4;           // 0: K=0,1   1: K=2,3
    const int l16  = lane & 15;

    const int arow = tile * 16 + l16;     // A row for this lane (same both halves)
    v8f c0 = {}, c1 = {}, c2 = {}, c3 = {};

#pragma unroll
    for (int k = 0; k < F_IN; k += 4) {
        // A fragment: 2 consecutive K values per lane
        const float* ap = X + (size_t)arow * F_IN + k + half * 2;
        v2f a; a[0] = ap[0]; a[1] = ap[1];

        // B fragments: col = Ntile*16 + l16, rows k+half*2 (+0,+1)
        const int krow = k + half * 2;
        const float* bp = W1 + (size_t)krow * HID + l16;
        v2f b0; b0[0] = bp[0];  b0[1] = bp[HID + 0];
        v2f b1; b1[0] = bp[16]; b1[1] = bp[HID + 16];
        v2f b2; b2[0] = bp[32]; b2[1] = bp[HID + 32];
        v2f b3; b3[0] = bp[48]; b3[1] = bp[HID + 48];

        c0 = __builtin_amdgcn_wmma_f32_16x16x4_f32(false, a, false, b0, (short)0, c0, false, false);
        c1 = __builtin_amdgcn_wmma_f32_16x16x4_f32(false, a, false, b1, (short)0, c1, false, false);
        c2 = __builtin_amdgcn_wmma_f32_16x16x4_f32(false, a, false, b2, (short)0, c2, false, false);
        c3 = __builtin_amdgcn_wmma_f32_16x16x4_f32(false, a, false, b3, (short)0, c3, false, false);
    }

    // C/D layout: reg r -> row r + half*8, col = l16 (+ N tile offset)
    float* hp = H + (size_t)(tile * 16 + half * 8) * HID + l16;
#pragma unroll
    for (int r = 0; r < 8; ++r) {
        hp[r * HID + 0]  = c0[r];
        hp[r * HID + 16] = c1[r];
        hp[r * HID + 32] = c2[r];
        hp[r * HID + 48] = c3[r];
    }
}

// ---------------- WMMA GEMM 2: H2[N,16] = Y[N,64] @ W2[64,16] ----------------

__global__ __launch_bounds__(32) void gemm2_wmma(const float* __restrict__ Y,
                                                 const float* __restrict__ W2,
                                                 float* __restrict__ H2) {
    const int tile = blockIdx.x;
    const int lane = threadIdx.x;
    const int half = lane >> 4;
    const int l16  = lane & 15;

    const int arow = tile * 16 + l16;
    v8f c0 = {};

#pragma unroll
    for (int k = 0; k < HID; k += 4) {
        const float* ap = Y + (size_t)arow * HID + k + half * 2;
        v2f a; a[0] = ap[0]; a[1] = ap[1];

        const int krow = k + half * 2;
        const float* bp = W2 + (size_t)krow * NCLS + l16;
        v2f b; b[0] = bp[0]; b[1] = bp[NCLS];

        c0 = __builtin_amdgcn_wmma_f32_16x16x4_f32(false, a, false, b, (short)0, c0, false, false);
    }

    float* hp = H2 + (size_t)(tile * 16 + half * 8) * NCLS + l16;
#pragma unroll
    for (int r = 0; r < 8; ++r) hp[r * NCLS] = c0[r];
}

// ---------------- edge scatter (layer 1): one wave per edge, 64 channels ----------------

__global__ void scatter64(const float* __restrict__ H,
                          const int* __restrict__ src, const int* __restrict__ dst,
                          const float* __restrict__ dis,
                          float* __restrict__ agg, int E) {
    int t = blockIdx.x * blockDim.x + threadIdx.x;
    int e = t >> 5;                 // edge id, one wave32 per edge
    int lane = t & 31;              // 2 channels per lane
    if (e >= E) return;
    int s = src[e], d = dst[e];
    float nrm = dis[s] * dis[d];
    const float2 hv = ((const float2*)(H + (size_t)s * HID))[lane];
    float* ap = agg + (size_t)d * HID + lane * 2;
    atomicAdd(ap + 0, hv.x * nrm);
    atomicAdd(ap + 1, hv.y * nrm);
}

// ---------------- edge scatter (layer 2): 16 channels per edge ----------------

__global__ void scatter16(const float* __restrict__ H2,
                          const int* __restrict__ src, const int* __restrict__ dst,
                          const float* __restrict__ dis,
                          float* __restrict__ out, int E) {
    int t = blockIdx.x * blockDim.x + threadIdx.x;
    int e = t >> 4;
    int j = t & 15;
    if (e >= E) return;
    int s = src[e], d = dst[e];
    float nrm = dis[s] * dis[d];
    atomicAdd(out + (size_t)d * NCLS + j, H2[(size_t)s * NCLS + j] * nrm);
}

// ---------------- launcher ----------------

extern "C" void kernel_launch(void* const* d_in, const int* in_sizes, int n_in,
                              void* d_out, int out_size, void* d_ws, size_t ws_size,
                              hipStream_t stream) {
    const float* x    = (const float*)d_in[0];
    const int*   esrc = (const int*)d_in[1];
    const int*   edst = (const int*)d_in[2];
    const float* W1   = (const float*)d_in[3];
    const float* b1   = (const float*)d_in[4];
    const float* W2   = (const float*)d_in[5];
    const float* b2   = (const float*)d_in[6];
    float* out = (float*)d_out;

    const int NN = in_sizes[0] / F_IN;     // 100000
    const int E  = in_sizes[1];            // 1,700,000
    const int Mtiles = NN / 16;            // 6250 (exact)

    // workspace layout
    float* dis = (float*)d_ws;                      // NN
    float* H   = dis + NN;                          // NN*64
    float* AGG = H + (size_t)NN * HID;              // NN*64 (also relu output)
    float* H2  = AGG + (size_t)NN * HID;            // NN*16

    const int B = 256;
    #define NB(n) (((n) + B - 1) / B)

    // degree -> dis
    zero_f32<<<NB(NN), B, 0, stream>>>(dis, NN);
    deg_count<<<NB(E), B, 0, stream>>>(edst, dis, E);
    deg_to_dis<<<NB(NN), B, 0, stream>>>(dis, NN);

    // layer 1: GEMM, init agg with bias, scatter, relu
    gemm1_wmma<<<Mtiles, 32, 0, stream>>>(x, W1, H);
    bias_broadcast<<<NB(NN * HID), B, 0, stream>>>(AGG, b1, NN * HID, HID - 1);
    scatter64<<<NB(E * 32), B, 0, stream>>>(H, esrc, edst, dis, AGG, E);
    relu_inplace<<<NB(NN * HID), B, 0, stream>>>(AGG, NN * HID);

    // layer 2: GEMM, init out with bias, scatter
    gemm2_wmma<<<Mtiles, 32, 0, stream>>>(AGG, W2, H2);
    bias_broadcast<<<NB(NN * NCLS), B, 0, stream>>>(out, b2, NN * NCLS, NCLS - 1);
    scatter16<<<NB(E * 16), B, 0, stream>>>(H2, esrc, edst, dis, out, E);

    #undef NB
}